// Attention_38998303047770
// MI455X (gfx1250) — compile-verified
//
#include <hip/hip_runtime.h>
#include <hip/hip_bf16.h>
#include <math.h>

#define N_HEADS 32
#define N_KV 8
#define HD 128
#define DIM 4096
#define KV_DIM 1024
#define BSZ 16
#define CACHE_LEN 4096
#define KVSTRIDE (N_KV * HD)   // 1024 floats between consecutive positions
#define NW 8

typedef float v2f __attribute__((ext_vector_type(2)));
typedef float v8f __attribute__((ext_vector_type(8)));

// D(16x16,f32) = A(16x4,f32) x B(4x16,f32) + C
// A layout: lanes 0-15 hold row M=lane, VGPR0=K0,VGPR1=K1; lanes 16-31 hold K2,K3
// B layout: lanes 0-15: VGPR0=row K0, VGPR1=row K1 (col=lane); lanes 16-31: rows K2,K3
// C/D layout: VGPR v = row v (lanes 0-15) / row v+8 (lanes 16-31), col = lane&15
__device__ __forceinline__ v8f wmma4(v2f a, v2f b, v8f c) {
#if __has_builtin(__builtin_amdgcn_wmma_f32_16x16x4_f32)
  return __builtin_amdgcn_wmma_f32_16x16x4_f32(false, a, false, b, (short)0, c,
                                               false, false);
#else
  // Layout-faithful cross-lane emulation (host parse / fallback only).
  const int lane = (int)(threadIdx.x & 31);
  const int l16 = lane & 15;
  const int hlf = lane >> 4;
  const float b0 = __shfl(b.x, l16, 32);
  const float b1 = __shfl(b.y, l16, 32);
  const float b2 = __shfl(b.x, l16 + 16, 32);
  const float b3 = __shfl(b.y, l16 + 16, 32);
#pragma unroll
  for (int v = 0; v < 8; v++) {
    const int row = v + hlf * 8;
    const float a0 = __shfl(a.x, row, 32);
    const float a1 = __shfl(a.y, row, 32);
    const float a2 = __shfl(a.x, row + 16, 32);
    const float a3 = __shfl(a.y, row + 16, 32);
    c[v] = fmaf(a0, b0, fmaf(a1, b1, fmaf(a2, b2, fmaf(a3, b3, c[v]))));
  }
  return c;
#endif
}

// One pipeline stage: 32 k-steps = 8 WMMA fragments.
struct Frag {
  v2f a[8];
  float bx[8], by[8];
};

// 16xK x Kx16 f32 GEMM tile. A row stride = DIM; W is (DIM x NCOLS) row-major,
// NCOLS compile-time so all stage loads use immediate offsets.
template <int NCOLS>
__device__ __forceinline__ v8f gemm16x16(const float* __restrict__ A,
                                         const float* __restrict__ W,
                                         int col0, int lane) {
  const int l16 = lane & 15;
  const int ksel = (lane >> 4) << 1;  // 0 or 2
  const float* arow = A + (size_t)l16 * DIM + ksel;
  const float* wbase = W + col0 + l16 + (size_t)ksel * NCOLS;

  auto stage = [&](Frag& f, int k) {
#pragma unroll
    for (int j = 0; j < 8; j++) {
      f.a[j] = *(const v2f*)(arow + k + 4 * j);
      f.bx[j] = wbase[(size_t)(k + 4 * j) * NCOLS];
      f.by[j] = wbase[(size_t)(k + 4 * j) * NCOLS + NCOLS];
    }
  };
  auto compute = [&](const Frag& f, v8f& acc) {
#pragma unroll
    for (int j = 0; j < 8; j++) {
      v2f b = {f.bx[j], f.by[j]};
      acc = wmma4(f.a[j], b, acc);
    }
  };

  v8f acc = {0.f, 0.f, 0.f, 0.f, 0.f, 0.f, 0.f, 0.f};
  Frag f0, f1;
  stage(f0, 0);
  for (int k0 = 0; k0 < DIM; k0 += 64) {
    stage(f1, k0 + 32);                 // DIM % 64 == 0 -> always in range
    if (k0 + 512 < DIM)
      __builtin_prefetch(wbase + (size_t)(k0 + 512) * NCOLS, 0, 1);
    compute(f0, acc);
    if (k0 + 64 < DIM) stage(f0, k0 + 64);
    compute(f1, acc);
  }
  return acc;
}

__device__ __forceinline__ void rope_apply(v8f& acc, int col,
                                           const float* __restrict__ fcos,
                                           const float* __restrict__ fsin) {
  const int d = col & (HD - 1);
  const float c = fcos[d >> 1];
  const float s = fsin[d >> 1];
#pragma unroll
  for (int v = 0; v < 8; v++) {
    float val = acc[v];
    float other = __shfl_xor(val, 1, 32);  // RoPE partner = adjacent column
    // even col holds x0: x0*c - x1*s ; odd col holds x1: x0*s + x1*c
    acc[v] = (d & 1) ? fmaf(other, s, val * c) : fmaf(val, c, -(other * s));
  }
}

template <int NCOLS>
__device__ __forceinline__ void store_tile(float* __restrict__ out, v8f acc,
                                           int col, int lane) {
  const int rbase = (lane >> 4) * 8;
#pragma unroll
  for (int v = 0; v < 8; v++)
    out[(size_t)(rbase + v) * NCOLS + col] = acc[v];
}

// ---------------- Kernel 1: fused QKV projection + RoPE -------------------
__global__ __launch_bounds__(128) void qkv_rope_kernel(
    const float* __restrict__ x, const float* __restrict__ wq,
    const float* __restrict__ wk, const float* __restrict__ wv,
    const float* __restrict__ fcos, const float* __restrict__ fsin,
    float* __restrict__ qo, float* __restrict__ ko, float* __restrict__ vo) {
  const int lane = threadIdx.x & 31;
  const int wave = threadIdx.x >> 5;
  const int n0 = (blockIdx.x * 4 + wave) * 16;
  const int l16 = lane & 15;

  if (n0 < DIM) {                       // Q projection
    v8f acc = gemm16x16<DIM>(x, wq, n0, lane);
    rope_apply(acc, n0 + l16, fcos, fsin);
    store_tile<DIM>(qo, acc, n0 + l16, lane);
  } else {                              // K or V projection
    const bool isK = n0 < DIM + KV_DIM;
    const float* W = isK ? wk : wv;
    float* out = isK ? ko : vo;
    const int col0 = n0 - DIM - (isK ? 0 : KV_DIM);
    v8f acc = gemm16x16<KV_DIM>(x, W, col0, lane);
    if (isK) rope_apply(acc, col0 + l16, fcos, fsin);
    store_tile<KV_DIM>(out, acc, col0 + l16, lane);
  }
}

// ---------------- Kernel 2: flash-decode attention ------------------------
__global__ __launch_bounds__(256) void attn_kernel(
    const float* __restrict__ xq, const float* __restrict__ xk,
    const float* __restrict__ xv, const float* __restrict__ cache_k,
    const float* __restrict__ cache_v, const float* __restrict__ shk,
    const float* __restrict__ shv, const int* __restrict__ sp_p,
    const int* __restrict__ spl_p, float* __restrict__ attn_out) {
  const int h = blockIdx.x;   // kv head
  const int b = blockIdx.y;   // batch
  const int lane = threadIdx.x & 31;
  const int wave = threadIdx.x >> 5;
  const int l16 = lane & 15;
  const int hlf = lane >> 4;
  const int ksel = hlf * 2;
  const int L = sp_p[0] + 1;      // total keys (incl. current token)
  const int spl = spl_p[0];       // shared prefix length

  // Per-(b,h) base pointers: row for key kidx is base + (idx << 10).
  const size_t hoff = (size_t)h * HD;
  const float* shk_h = shk + hoff;
  const float* shv_h = shv + hoff;
  const float* ck_h = cache_k + (size_t)b * CACHE_LEN * KVSTRIDE + hoff;
  const float* cv_h = cache_v + (size_t)b * CACHE_LEN * KVSTRIDE + hoff;
  const float* xk_h = xk + (size_t)b * KVSTRIDE + hoff;
  const float* xv_h = xv + (size_t)b * KVSTRIDE + hoff;

  __shared__ float qsh[16][132];          // padded: avoid bank conflicts
  __shared__ float pbuf[NW][16][18];      // P-tile transpose scratch (per wave)
  __shared__ float obuf[NW][4][HD];
  __shared__ float mbuf[NW][4];
  __shared__ float lbuf[NW][4];

  // Q (4 valid heads) -> LDS, pre-scaled, rows 4..15 zeroed (keeps S clean)
  const float qscale = 0.088388347648318447f;  // 1/sqrt(128)
  for (int idx = threadIdx.x; idx < 16 * HD; idx += 256) {
    const int r = idx >> 7, d = idx & (HD - 1);
    float qv = 0.f;
    if (r < 4) qv = xq[((size_t)b * N_HEADS + h * 4 + r) * HD + d] * qscale;
    qsh[r][d] = qv;
  }
  __syncthreads();

  v8f o[8];
#pragma unroll
  for (int dt = 0; dt < 8; dt++) o[dt] = (v8f){0.f,0.f,0.f,0.f,0.f,0.f,0.f,0.f};
  float m_run[8], l_run[8];
#pragma unroll
  for (int v = 0; v < 8; v++) { m_run[v] = -3.0e38f; l_run[v] = 0.f; }

  const int ntiles = (L + 15) >> 4;
  for (int t = wave; t < ntiles; t += NW) {
    const int kb = t << 4;

    // ---- S tile = Q x K^T (rows = q-heads padded to 16, cols = 16 keys) ----
    const int mykey = kb + l16;  // this lane's key row for the B operand
    const float* krow =
        (mykey < spl)    ? shk_h + ((size_t)mykey << 10)
        : (mykey < L - 1) ? ck_h + ((size_t)(mykey - spl) << 10)
                          : xk_h;  // current token / OOB fallback
    krow += ksel;

    v8f st = {0.f, 0.f, 0.f, 0.f, 0.f, 0.f, 0.f, 0.f};
#pragma unroll
    for (int kd = 0; kd < HD; kd += 4) {
      v2f a = *(const v2f*)&qsh[l16][kd + ksel];
      v2f bb = *(const v2f*)(krow + kd);
      st = wmma4(a, bb, st);
    }
    if (mykey >= L) {
#pragma unroll
      for (int v = 0; v < 8; v++) st[v] = -1.0e30f;  // mask invalid key columns
    }

    // ---- online softmax (row = vgpr, reduce over 16 key lanes) ----
    float osc[8];
#pragma unroll
    for (int v = 0; v < 8; v++) {
      float m = st[v];
      m = fmaxf(m, __shfl_xor(m, 1, 32));
      m = fmaxf(m, __shfl_xor(m, 2, 32));
      m = fmaxf(m, __shfl_xor(m, 4, 32));
      m = fmaxf(m, __shfl_xor(m, 8, 32));
      const float mnew = fmaxf(m_run[v], m);
      const float sc = __expf(m_run[v] - mnew);
      float p = __expf(st[v] - mnew);
      float rs = p;
      rs += __shfl_xor(rs, 1, 32);
      rs += __shfl_xor(rs, 2, 32);
      rs += __shfl_xor(rs, 4, 32);
      rs += __shfl_xor(rs, 8, 32);
      l_run[v] = l_run[v] * sc + rs;
      m_run[v] = mnew;
      osc[v] = sc;
      st[v] = p;
    }
#pragma unroll
    for (int dt = 0; dt < 8; dt++)
#pragma unroll
      for (int v = 0; v < 8; v++) o[dt][v] = o[dt][v] * osc[v];

    // ---- transpose P (D-layout -> A-layout) through wave-local LDS ----
#pragma unroll
    for (int v = 0; v < 8; v++) pbuf[wave][hlf * 8 + v][l16] = st[v];
    __builtin_amdgcn_wave_barrier();
    v2f pa[4];
#pragma unroll
    for (int kc = 0; kc < 4; kc++)
      pa[kc] = *(const v2f*)&pbuf[wave][l16][kc * 4 + ksel];
    __builtin_amdgcn_wave_barrier();

    // ---- O += P x V ----
#pragma unroll
    for (int kc = 0; kc < 4; kc++) {
      const int key0 = kb + kc * 4 + ksel;
      const int key1 = key0 + 1;
      const float* vr0 =
          (key0 < spl)    ? shv_h + ((size_t)key0 << 10)
          : (key0 < L - 1) ? cv_h + ((size_t)(key0 - spl) << 10)
                           : xv_h;
      const float* vr1 =
          (key1 < spl)    ? shv_h + ((size_t)key1 << 10)
          : (key1 < L - 1) ? cv_h + ((size_t)(key1 - spl) << 10)
                           : xv_h;
#pragma unroll
      for (int dt = 0; dt < 8; dt++) {
        v2f bb;
        bb.x = vr0[dt * 16 + l16];
        bb.y = vr1[dt * 16 + l16];
        o[dt] = wmma4(pa[kc], bb, o[dt]);
      }
    }
  }

  // ---- per-wave partials -> LDS ----
  if (hlf == 0) {
#pragma unroll
    for (int v = 0; v < 4; v++) {
      if (l16 == 0) { mbuf[wave][v] = m_run[v]; lbuf[wave][v] = l_run[v]; }
#pragma unroll
      for (int dt = 0; dt < 8; dt++) obuf[wave][v][dt * 16 + l16] = o[dt][v];
    }
  }
  __syncthreads();

  // ---- log-sum-exp combine across the 8 waves ----
  for (int idx = threadIdx.x; idx < 4 * HD; idx += 256) {
    const int qh = idx >> 7, d = idx & (HD - 1);
    float M = -3.0e38f;
#pragma unroll
    for (int w = 0; w < NW; w++) M = fmaxf(M, mbuf[w][qh]);
    float num = 0.f, den = 0.f;
#pragma unroll
    for (int w = 0; w < NW; w++) {
      const float f = __expf(mbuf[w][qh] - M);
      num += obuf[w][qh][d] * f;
      den += lbuf[w][qh] * f;
    }
    attn_out[((size_t)b * N_HEADS + h * 4 + qh) * HD + d] = num / den;
  }
}

// ---------------- Kernel 3: output projection -----------------------------
__global__ __launch_bounds__(128) void out_proj_kernel(
    const float* __restrict__ attn, const float* __restrict__ wo,
    float* __restrict__ out) {
  const int lane = threadIdx.x & 31;
  const int wave = threadIdx.x >> 5;
  const int col0 = (blockIdx.x * 4 + wave) * 16;
  v8f acc = gemm16x16<DIM>(attn, wo, col0, lane);
  store_tile<DIM>(out, acc, col0 + (lane & 15), lane);
}

extern "C" void kernel_launch(void* const* d_in, const int* in_sizes, int n_in,
                              void* d_out, int out_size, void* d_ws,
                              size_t ws_size, hipStream_t stream) {
  const float* x   = (const float*)d_in[0];
  const float* wq  = (const float*)d_in[1];
  const float* wk  = (const float*)d_in[2];
  const float* wv  = (const float*)d_in[3];
  const float* wo  = (const float*)d_in[4];
  const float* ck  = (const float*)d_in[5];
  const float* cv  = (const float*)d_in[6];
  const float* shk = (const float*)d_in[7];
  const float* shv = (const float*)d_in[8];
  const float* fc  = (const float*)d_in[9];
  const float* fs  = (const float*)d_in[10];
  const int* sp    = (const int*)d_in[11];
  const int* spl   = (const int*)d_in[12];

  float* ws = (float*)d_ws;
  float* q    = ws;                               // 16*4096
  float* k    = q + (size_t)BSZ * DIM;            // 16*1024
  float* v    = k + (size_t)BSZ * KV_DIM;         // 16*1024
  float* attn = v + (size_t)BSZ * KV_DIM;         // 16*4096

  // 384 column tiles (Q:256, K:64, V:64), 4 waves (tiles) per block
  qkv_rope_kernel<<<(DIM + 2 * KV_DIM) / 16 / 4, 128, 0, stream>>>(
      x, wq, wk, wv, fc, fs, q, k, v);
  // one block per (kv head, batch), 8 waves split the key tiles
  attn_kernel<<<dim3(N_KV, BSZ), 256, 0, stream>>>(q, k, v, ck, cv, shk, shv,
                                                   sp, spl, attn);
  // 256 column tiles, 4 per block
  out_proj_kernel<<<DIM / 16 / 4, 128, 0, stream>>>(attn, wo, (float*)d_out);
}